// GCN_test_v2_2190433321523
// MI455X (gfx1250) — compile-verified
//
#include <hip/hip_runtime.h>
#include <hip/hip_bf16.h>

// ---------------------------------------------------------------------------
// GCN 2-layer forward for MI455X (gfx1250).
//   h1  = x @ W1                         (WMMA f32 16x16x4, W1 staged in LDS)
//   agg = segment_sum(w_e * h1[src] -> dst)   (fp32 global atomics, L2-resident)
//   h2  = relu(agg) @ W2                 (WMMA f32 16x16x4, relu fused)
//   out = segment_sum(w_e * h2[src] -> dst)
// ---------------------------------------------------------------------------

typedef __attribute__((ext_vector_type(2))) float v2f;
typedef __attribute__((ext_vector_type(8))) float v8f;

#define NFEAT 128
#define NHID  64
#define NCLS  16

static __device__ __forceinline__ v8f wmma_f32(v2f a, v2f b, v8f c) {
    // V_WMMA_F32_16X16X4_F32 : D(16x16 f32) = A(16x4) * B(4x16) + C
    return __builtin_amdgcn_wmma_f32_16x16x4_f32(
        /*neg_a=*/false, a, /*neg_b=*/false, b,
        /*c_mod=*/(short)0, c, /*reuse_a=*/false, /*reuse_b=*/false);
}

// -------------------- GEMM1: [N,128] x [128,64] -> h1 [N,64] ----------------
// B fragments for all 32 k-steps x 4 n-tiles staged once per block into LDS
// in WMMA-native layout: ldsB[(kstep*4 + tile)*32 + lane] = {W1[k,col], W1[k+1,col]}
__global__ void __launch_bounds__(256)
gcn_gemm1(const float* __restrict__ x,
          const float* __restrict__ W1,
          float* __restrict__ h1,
          int n_tiles) {
    __shared__ v2f ldsB[32 * 4 * 32];            // 32 KB

    // ---- cooperative stage of W1 (fragment layout) ----
    for (int idx = threadIdx.x; idx < 32 * 4 * 32; idx += 256) {
        const int lane = idx & 31;
        const int t    = (idx >> 5) & 3;         // n-tile
        const int ks   = idx >> 7;               // k-step
        const int krow = ks * 4 + 2 * (lane >> 4);
        const int col  = t * 16 + (lane & 15);
        v2f b = { W1[(long)krow * NHID + col], W1[(long)(krow + 1) * NHID + col] };
        ldsB[idx] = b;
    }
    __syncthreads();

    const int wave = blockIdx.x * (blockDim.x >> 5) + (threadIdx.x >> 5);
    if (wave < n_tiles) {                        // wave-uniform: EXEC all-ones
        const int lane = threadIdx.x & 31;
        const int m    = lane & 15;
        const int kh   = lane >> 4;
        const long rowbase = (long)wave * 16;

        const float* ap = x + (rowbase + m) * NFEAT + 2 * kh;

        v8f acc0 = {}, acc1 = {}, acc2 = {}, acc3 = {};
        #pragma unroll 4
        for (int ks = 0; ks < 32; ++ks) {
            v2f a = *(const v2f*)(ap + ks * 4);
            const v2f* bl = &ldsB[ks * 128 + lane];
            acc0 = wmma_f32(a, bl[ 0], acc0);    // ds_load_b64 per fragment
            acc1 = wmma_f32(a, bl[32], acc1);
            acc2 = wmma_f32(a, bl[64], acc2);
            acc3 = wmma_f32(a, bl[96], acc3);
        }

        // C layout: VGPR r, lanes 0-15 -> M=r, lanes 16-31 -> M=r+8, N=lane&15
        float* op = h1 + (rowbase + 8 * kh) * NHID + m;
        #pragma unroll
        for (int r = 0; r < 8; ++r) {
            op[(long)r * NHID +  0] = acc0[r];
            op[(long)r * NHID + 16] = acc1[r];
            op[(long)r * NHID + 32] = acc2[r];
            op[(long)r * NHID + 48] = acc3[r];
        }
    }
}

// -------------------- GEMM2: relu(agg)[N,64] x [64,16] -> h2 [N,16] ---------
__global__ void __launch_bounds__(256)
gcn_gemm2(const float* __restrict__ agg,
          const float* __restrict__ W2,
          float* __restrict__ h2,
          int n_tiles) {
    __shared__ v2f ldsB[16 * 32];                // 4 KB: 16 k-steps, 1 n-tile

    for (int idx = threadIdx.x; idx < 16 * 32; idx += 256) {
        const int lane = idx & 31;
        const int ks   = idx >> 5;
        const int krow = ks * 4 + 2 * (lane >> 4);
        const int col  = lane & 15;
        v2f b = { W2[(long)krow * NCLS + col], W2[(long)(krow + 1) * NCLS + col] };
        ldsB[idx] = b;
    }
    __syncthreads();

    const int wave = blockIdx.x * (blockDim.x >> 5) + (threadIdx.x >> 5);
    if (wave < n_tiles) {
        const int lane = threadIdx.x & 31;
        const int m    = lane & 15;
        const int kh   = lane >> 4;
        const long rowbase = (long)wave * 16;

        const float* ap = agg + (rowbase + m) * NHID + 2 * kh;

        v8f acc = {};
        #pragma unroll 4
        for (int ks = 0; ks < 16; ++ks) {
            v2f a = *(const v2f*)(ap + ks * 4);
            a.x = fmaxf(a.x, 0.0f);              // fused ReLU
            a.y = fmaxf(a.y, 0.0f);
            acc = wmma_f32(a, ldsB[ks * 32 + lane], acc);
        }

        float* op = h2 + (rowbase + 8 * kh) * NCLS + m;
        #pragma unroll
        for (int r = 0; r < 8; ++r)
            op[(long)r * NCLS] = acc[r];
    }
}

// -------------------- edge scatter (layer 1): 64 feats, float4 chunks -------
__global__ void gcn_scatter1(const long long* __restrict__ ei,  // [2,E]
                             const float* __restrict__ w,
                             const float* __restrict__ h1,
                             float* __restrict__ agg,
                             int E) {
    const int i = blockIdx.x * blockDim.x + threadIdx.x;   // E * 16 items
    if (i >= E * 16) return;
    const int e  = i >> 4;
    const int f4 = i & 15;
    const long s = (long)ei[e];
    const long d = (long)ei[(long)E + e];
    const float we = w[e];
    const float4 v = ((const float4*)(h1 + s * NHID))[f4];
    float* outp = agg + d * NHID + f4 * 4;
    atomicAdd(outp + 0, we * v.x);
    atomicAdd(outp + 1, we * v.y);
    atomicAdd(outp + 2, we * v.z);
    atomicAdd(outp + 3, we * v.w);
}

// -------------------- edge scatter (layer 2): 16 feats ----------------------
__global__ void gcn_scatter2(const long long* __restrict__ ei,
                             const float* __restrict__ w,
                             const float* __restrict__ h2,
                             float* __restrict__ out,
                             int E) {
    const int i = blockIdx.x * blockDim.x + threadIdx.x;   // E * 4 items
    if (i >= E * 4) return;
    const int e  = i >> 2;
    const int f4 = i & 3;
    const long s = (long)ei[e];
    const long d = (long)ei[(long)E + e];
    const float we = w[e];
    const float4 v = ((const float4*)(h2 + s * NCLS))[f4];
    float* outp = out + d * NCLS + f4 * 4;
    atomicAdd(outp + 0, we * v.x);
    atomicAdd(outp + 1, we * v.y);
    atomicAdd(outp + 2, we * v.z);
    atomicAdd(outp + 3, we * v.w);
}

// -------------------- zero fill ---------------------------------------------
__global__ void gcn_zero(float* __restrict__ p, long n) {
    long i = (long)blockIdx.x * blockDim.x + threadIdx.x;
    const long stride = (long)gridDim.x * blockDim.x;
    for (; i < n; i += stride) p[i] = 0.0f;
}

extern "C" void kernel_launch(void* const* d_in, const int* in_sizes, int n_in,
                              void* d_out, int out_size, void* d_ws, size_t ws_size,
                              hipStream_t stream) {
    const float*     x  = (const float*)d_in[0];
    const long long* ei = (const long long*)d_in[1];   // int64 [2,E]
    const float*     w  = (const float*)d_in[2];
    const float*     W1 = (const float*)d_in[3];
    const float*     W2 = (const float*)d_in[4];
    float* out = (float*)d_out;

    const int N = in_sizes[0] / NFEAT;   // 100000
    const int E = in_sizes[2];           // 1600000

    // workspace: h1 [N,64] | agg [N,64] | h2 [N,16]
    float* h1  = (float*)d_ws;
    float* agg = h1  + (long)N * NHID;
    float* h2  = agg + (long)N * NHID;

    const int n_tiles = N / 16;          // 6250 (exact)
    const int waves_per_blk = 8;         // 256 threads (wave32)
    const int gemm_blocks = (n_tiles + waves_per_blk - 1) / waves_per_blk;

    // ---- layer 1 ----
    gcn_gemm1<<<gemm_blocks, 256, 0, stream>>>(x, W1, h1, n_tiles);
    gcn_zero<<<1024, 256, 0, stream>>>(agg, (long)N * NHID);
    {
        const long items = (long)E * 16;
        const int blocks = (int)((items + 255) / 256);
        gcn_scatter1<<<blocks, 256, 0, stream>>>(ei, w, h1, agg, E);
    }

    // ---- layer 2 ----
    gcn_gemm2<<<gemm_blocks, 256, 0, stream>>>(agg, W2, h2, n_tiles);
    gcn_zero<<<1024, 256, 0, stream>>>(out, (long)N * NCLS);
    {
        const long items = (long)E * 4;
        const int blocks = (int)((items + 255) / 256);
        gcn_scatter2<<<blocks, 256, 0, stream>>>(ei, w, h2, out, E);
    }
}